// FkMigrationLinear_68848325755339
// MI455X (gfx1250) — compile-verified
//
#include <hip/hip_runtime.h>
#include <hip/hip_bf16.h>
#include <math.h>

// ---------------- geometry constants (match reference) ----------------
#define NB      4
#define NDET    128
#define NT      1024
#define NX      256
#define NY      256
#define NF      513          // rfft bins of 1024
#define NFP     528          // NF padded to multiple of 16 (zero pad)
#define TWO_NFP (2*NFP)      // spec stored as [re | im] planes of width NFP

typedef __attribute__((ext_vector_type(2))) float v2f;
typedef __attribute__((ext_vector_type(8))) float v8f;

static __device__ __forceinline__ v8f wmma4(v2f a, v2f b, v8f c) {
  // D = A(16x4,f32) * B(4x16,f32) + C(16x16,f32)
  return __builtin_amdgcn_wmma_f32_16x16x4_f32(false, a, false, b, (short)0, c,
                                               false, false);
}
static __device__ __forceinline__ v8f zero8() {
  v8f z = {0.f,0.f,0.f,0.f,0.f,0.f,0.f,0.f};
  return z;
}

// ---------------- workspace layout (float offsets) ----------------
// stats[0..3]=mean[b], stats[4..7]=inv_std[b], stats[8]=sum(freq_weight)
#define OFF_STATS 0
#define OFF_XW    16                          // 512*1024            = 524288
#define OFF_DTT   (OFF_XW + 524288)           // 1024*1056           = 1081344
#define OFF_SPEC  (OFF_DTT + 1081344)         // 512*1056            = 540672
#define OFF_WRE   (OFF_SPEC + 540672)         // 128*128             = 16384
#define OFF_WIM   (OFF_WRE + 16384)           // 128*128             = 16384
#define OFF_SKFR  (OFF_WIM + 16384)           // 4*128*528           = 270336
#define OFF_SKFI  (OFF_SKFR + 270336)         // 4*128*528           = 270336
#define OFF_TMPA  (OFF_SKFI + 270336)         // 1024*256            = 262144
#define OFF_BX    (OFF_TMPA + 262144)         // 256*256             = 65536

// ---------------- K0: per-batch mean / inv_std, and sum(freq_weight) -----
__global__ void fk_stats(const float* __restrict__ sino,
                         const float* __restrict__ fw,
                         float* __restrict__ stats) {
  __shared__ double sh1[256];
  __shared__ double sh2[256];
  const int b = blockIdx.x, tid = threadIdx.x;
  if (b < NB) {
    const float* p = sino + b * (NDET * NT);
    double s1 = 0.0, s2 = 0.0;
    for (int i = tid; i < NDET * NT; i += 256) {
      double v = (double)p[i];
      s1 += v; s2 += v * v;
    }
    sh1[tid] = s1; sh2[tid] = s2; __syncthreads();
    for (int s = 128; s > 0; s >>= 1) {
      if (tid < s) { sh1[tid] += sh1[tid + s]; sh2[tid] += sh2[tid + s]; }
      __syncthreads();
    }
    if (tid == 0) {
      double n = (double)(NDET * NT);
      double mean = sh1[0] / n;
      double var  = sh2[0] / n - mean * mean;
      stats[b]     = (float)mean;
      stats[4 + b] = (float)(1.0 / sqrt(var + 1.1920928955078125e-7));
    }
  } else {
    double s = 0.0;
    for (int i = tid; i < NF; i += 256) s += (double)fw[i];
    sh1[tid] = s; __syncthreads();
    for (int st = 128; st > 0; st >>= 1) {
      if (tid < st) sh1[tid] += sh1[tid + st];
      __syncthreads();
    }
    if (tid == 0) stats[8] = (float)sh1[0];
  }
}

// ---------------- K1: normalized + windowed input xw[(b*128+d)][t] -------
__global__ void fk_xw(const float* __restrict__ sino,
                      const float* __restrict__ tw,
                      const float* __restrict__ apod,
                      const float* __restrict__ stats,
                      float* __restrict__ xw) {
  int idx = blockIdx.x * blockDim.x + threadIdx.x;
  if (idx >= NB * NDET * NT) return;
  int t = idx & (NT - 1);
  int d = (idx >> 10) & (NDET - 1);
  int b = idx >> 17;
  float v = (sino[idx] - stats[b]) * stats[4 + b];
  xw[idx] = v * tw[t] * apod[d];
}

// ---------------- K2: time-DFT table [1024][2*528]: cos | -sin -----------
__global__ void fk_dtt(float* __restrict__ dtt) {
  int idx = blockIdx.x * blockDim.x + threadIdx.x;
  if (idx >= NT * TWO_NFP) return;
  int col = idx % TWO_NFP;
  int t   = idx / TWO_NFP;
  int f   = (col < NFP) ? col : col - NFP;
  float val = 0.f;
  if (f < NF) {
    int ph = (f * t) & (NT - 1);            // exact integer phase
    float ang = 6.283185307179586f * (float)ph / (float)NT;
    float s, c; __sincosf(ang, &s, &c);
    val = (col < NFP) ? c : -s;
  }
  dtt[idx] = val;
}

// ---------------- K3: detector-DFT twiddles e^{-2pi i k d /128} ----------
__global__ void fk_wtab(float* __restrict__ wre, float* __restrict__ wim) {
  int idx = blockIdx.x * blockDim.x + threadIdx.x;
  if (idx >= NDET * NDET) return;
  int d  = idx & (NDET - 1);
  int kx = idx >> 7;
  int ph = (kx * d) & (NDET - 1);
  float ang = 6.283185307179586f * (float)ph / (float)NDET;
  float s, c; __sincosf(ang, &s, &c);
  wre[idx] = c;
  wim[idx] = -s;
}

// ---------------- K4: rfft as WMMA GEMM: spec = xw(512x1024) * dtt -------
__global__ __launch_bounds__(32)
void fk_specgemm(const float* __restrict__ xw,
                 const float* __restrict__ dtt,
                 float* __restrict__ spec) {
  const int m0 = blockIdx.x * 16, n0 = blockIdx.y * 16;
  const int lane = threadIdx.x, half = lane >> 4, r = lane & 15;
  v8f acc = zero8();
  const float* arow = xw + (m0 + r) * NT;
  for (int k0 = 0; k0 < NT; k0 += 4) {
    const int kb = k0 + half * 2;
    v2f a  = { arow[kb], arow[kb + 1] };
    v2f bm = { dtt[kb * TWO_NFP + n0 + r], dtt[(kb + 1) * TWO_NFP + n0 + r] };
    acc = wmma4(a, bm, acc);
  }
  for (int rr = 0; rr < 8; ++rr) {
    int m = m0 + rr + half * 8;
    spec[m * TWO_NFP + n0 + r] = acc[rr];
  }
}

// ---------------- K5: detector fft + evanescent mask + weights -----------
__global__ void fk_kxdft(const float* __restrict__ spec,
                         const float* __restrict__ wre,
                         const float* __restrict__ wim,
                         const float* __restrict__ prop_mask,
                         const float* __restrict__ fw,
                         float* __restrict__ skr,
                         float* __restrict__ ski) {
  int idx = blockIdx.x * blockDim.x + threadIdx.x;
  if (idx >= NB * NDET * NFP) return;
  const int fp = idx % NFP;
  const int t1 = idx / NFP;          // b*128 + kx
  const int kx = t1 & (NDET - 1);
  const int b  = t1 >> 7;
  float re = 0.f, im = 0.f;
  if (fp < NF) {
    const float* sp = spec + (b * NDET) * TWO_NFP;
    const float* wr = wre + kx * NDET;
    const float* wi = wim + kx * NDET;
    for (int d = 0; d < NDET; ++d) {
      float sr = sp[d * TWO_NFP + fp];
      float si = sp[d * TWO_NFP + NFP + fp];
      float cr = wr[d], ci = wi[d];
      re = fmaf(sr, cr, fmaf(-si, ci, re));
      im = fmaf(sr, ci, fmaf( si, cr, im));
    }
    float mw = prop_mask[kx * NF + fp] * fw[fp];
    re *= mw; im *= mw;
  }
  skr[idx] = re;   // zero in pad region keeps later K-loops clean
  ski[idx] = im;
}

// ---------------- K6: streams phase_y once; complex contraction over f ---
// tmp[b,y,d] = sum_f skf[b,d,f] * phase_y[y,d,f]; written packed for K8:
// tmpA[(b*NY+y)][k] with k<128 -> re(d=k), k>=128 -> im(d=k-128)
__global__ __launch_bounds__(32)
void fk_ycontract(const float* __restrict__ py,     // complex interleaved
                  const float* __restrict__ skr,
                  const float* __restrict__ ski,
                  float* __restrict__ tmpA) {
  const int d = blockIdx.x;
  const int y0 = blockIdx.y * 16;
  const int lane = threadIdx.x, half = lane >> 4, r = lane & 15;
  v8f arr = zero8(), aii = zero8(), ari = zero8(), air = zero8();
  const int arow = ((y0 + r) * NDET + d) * NF;      // phase_y row base (complex idx)
  const int sb0  = (r * NDET + d) * NFP;            // skf row for batch r (r<4)
  for (int k0 = 0; k0 < 516; k0 += 4) {
    const int f = k0 + half * 2;
    float re0 = 0.f, im0 = 0.f, re1 = 0.f, im1 = 0.f;
    if (f + 1 < NF) {
      const float2 p0 = *(const float2*)(py + (arow + f) * 2);
      const float2 p1 = *(const float2*)(py + (arow + f + 1) * 2);
      re0 = p0.x; im0 = p0.y; re1 = p1.x; im1 = p1.y;
    } else if (f < NF) {
      re0 = py[(arow + f) * 2];
      im0 = py[(arow + f) * 2 + 1];
    }
    v2f are = { re0, re1 }, aim = { im0, im1 };
    float br0 = 0.f, br1 = 0.f, bi0 = 0.f, bi1 = 0.f;
    if (r < NB) {                                   // B columns 0..3 = batches
      br0 = skr[sb0 + f]; br1 = skr[sb0 + f + 1];   // pad region is zero
      bi0 = ski[sb0 + f]; bi1 = ski[sb0 + f + 1];
    }
    v2f bre = { br0, br1 }, bim = { bi0, bi1 };
    arr = wmma4(are, bre, arr);
    aii = wmma4(aim, bim, aii);
    ari = wmma4(are, bim, ari);
    air = wmma4(aim, bre, air);
  }
  if (r < NB) {
    const int b = r;
    for (int rr = 0; rr < 8; ++rr) {
      const int y = y0 + rr + half * 8;
      const int o = (b * NY + y) * 256;
      tmpA[o + d]       = arr[rr] - aii[rr];        // re
      tmpA[o + 128 + d] = ari[rr] + air[rr];        // im
    }
  }
}

// ---------------- K7: pack Bx = [px_re ; -px_im]  (256 x 256) ------------
__global__ void fk_bx(const float* __restrict__ px, float* __restrict__ bx) {
  int idx = blockIdx.x * blockDim.x + threadIdx.x;
  if (idx >= 256 * NX) return;
  int x = idx & (NX - 1);
  int k = idx >> 8;
  bx[idx] = (k < NDET) ? px[(k * NX + x) * 2]
                       : -px[((k - NDET) * NX + x) * 2 + 1];
}

// ---------------- K8: img = Re part via real GEMM (1024 x 256 x K=256) ---
__global__ __launch_bounds__(32)
void fk_img(const float* __restrict__ tmpA,
            const float* __restrict__ bx,
            const float* __restrict__ stats,
            float* __restrict__ out) {
  const int m0 = blockIdx.x * 16, x0 = blockIdx.y * 16;
  const int lane = threadIdx.x, half = lane >> 4, r = lane & 15;
  v8f acc = zero8();
  const float* arow = tmpA + (m0 + r) * 256;
  for (int k0 = 0; k0 < 256; k0 += 4) {
    const int kb = k0 + half * 2;
    v2f a  = { arow[kb], arow[kb + 1] };
    v2f bm = { bx[kb * NX + x0 + r], bx[(kb + 1) * NX + x0 + r] };
    acc = wmma4(a, bm, acc);
  }
  const float scale = 1.0f / ((float)NDET * stats[8]);   // 1/(128*sum(w))
  for (int rr = 0; rr < 8; ++rr) {
    const int m = m0 + rr + half * 8;                    // m = b*NY + y
    out[m * NX + x0 + r] = acc[rr] * scale;
  }
}

// ---------------- host-side launch -----------------
extern "C" void kernel_launch(void* const* d_in, const int* in_sizes, int n_in,
                              void* d_out, int out_size, void* d_ws, size_t ws_size,
                              hipStream_t stream) {
  const float* sino = (const float*)d_in[0];
  const float* tw   = (const float*)d_in[1];
  const float* apod = (const float*)d_in[2];
  const float* fw   = (const float*)d_in[3];
  const float* pmsk = (const float*)d_in[4];
  const float* px   = (const float*)d_in[5];   // complex64 interleaved
  const float* py   = (const float*)d_in[6];   // complex64 interleaved
  float* out = (float*)d_out;

  float* ws    = (float*)d_ws;
  float* stats = ws + OFF_STATS;
  float* xw    = ws + OFF_XW;
  float* dtt   = ws + OFF_DTT;
  float* spec  = ws + OFF_SPEC;
  float* wre   = ws + OFF_WRE;
  float* wim   = ws + OFF_WIM;
  float* skr   = ws + OFF_SKFR;
  float* ski   = ws + OFF_SKFI;
  float* tmpA  = ws + OFF_TMPA;
  float* bxb   = ws + OFF_BX;

  fk_stats<<<5, 256, 0, stream>>>(sino, fw, stats);
  fk_xw<<<(NB * NDET * NT + 255) / 256, 256, 0, stream>>>(sino, tw, apod, stats, xw);
  fk_dtt<<<(NT * TWO_NFP + 255) / 256, 256, 0, stream>>>(dtt);
  fk_wtab<<<(NDET * NDET + 255) / 256, 256, 0, stream>>>(wre, wim);
  fk_specgemm<<<dim3(32, 66), 32, 0, stream>>>(xw, dtt, spec);
  fk_kxdft<<<(NB * NDET * NFP + 255) / 256, 256, 0, stream>>>(spec, wre, wim, pmsk, fw, skr, ski);
  fk_ycontract<<<dim3(NDET, 16), 32, 0, stream>>>(py, skr, ski, tmpA);
  fk_bx<<<(256 * NX + 255) / 256, 256, 0, stream>>>(px, bxb);
  fk_img<<<dim3(64, 16), 32, 0, stream>>>(tmpA, bxb, stats, out);
}